// Encoding3D_90125593740048
// MI455X (gfx1250) — compile-verified
//
#include <hip/hip_runtime.h>
#include <stdint.h>

// ---------------------------------------------------------------------------
// Encoding3D fused kernel for MI455X (gfx1250, wave32).
//   stage X/codewords via GLOBAL_LOAD_ASYNC_TO_LDS_B128 (ASYNCcnt path)
//   GEMM1 (X·Cᵀ) -> scaled-L2 logits -> softmax(K=32) -> coefA store (NT)
//   GEMM2 (Aᵀ·X) -> atomic partial AtX -> finalize E = AtX - Asum·C
// Both GEMMs use V_WMMA_F32_16X16X4_F32 (f32 end-to-end; problem is
// HBM-bandwidth-bound at ~192MB / 23.3TB/s, so precision costs nothing).
// ---------------------------------------------------------------------------

typedef float v2f __attribute__((ext_vector_type(2)));
typedef float v8f __attribute__((ext_vector_type(8)));

#ifndef __has_builtin
#define __has_builtin(x) 0
#endif
#if __has_builtin(__builtin_amdgcn_wmma_f32_16x16x4_f32)
#define HAS_WMMA_F32 1
#else
#define HAS_WMMA_F32 0
#endif

constexpr int Bsz = 4;
constexpr int Csz = 64;
constexpr int Ksz = 32;
constexpr int Nsz = 32 * 64 * 64;      // 131072
constexpr int TILE_N = 128;            // n-rows per workgroup
constexpr int XS_STRIDE = 132;         // [c][n] pad: bank stride 4 -> conflict-free frags
constexpr int CS_STRIDE = 68;          // [k][c] pad
constexpr int AS_STRIDE = 133;         // [k][n] pad: odd stride -> conflict-free
constexpr int WS_ATX = Bsz * Ksz * Csz;      // 8192 floats
constexpr int WS_TOTAL = WS_ATX + Bsz * Ksz; // + Asum

// Async memory->LDS copy of 16B per lane (512B per wave instruction).
// ldsOff: byte offset within the workgroup LDS allocation (low 32 bits of a
// generic pointer to __shared__, per ISA aperture mapping). Tracked by ASYNCcnt.
__device__ __forceinline__ void async_copy_b128(const void* gsrc, void* ldsDst) {
    unsigned ldsOff = (unsigned)(uintptr_t)ldsDst;
    unsigned long long gaddr = (unsigned long long)(uintptr_t)gsrc;
    asm volatile("global_load_async_to_lds_b128 %0, %1, off"
                 :: "v"(ldsOff), "v"(gaddr) : "memory");
}
__device__ __forceinline__ void async_wait_all() {
    asm volatile("s_wait_asynccnt 0x0" ::: "memory");
}

__global__ void enc3d_init_ws(float* ws, int n) {
    int i = blockIdx.x * blockDim.x + threadIdx.x;
    if (i < n) ws[i] = 0.0f;
}

__global__ __launch_bounds__(256)
void enc3d_main(const float* __restrict__ X,      // [B][C][N]
                const float* __restrict__ CW,     // [K][C]
                const float* __restrict__ scale,  // [K]
                float* __restrict__ out,          // [E(8192) | coefA(B*K*N)]
                float* __restrict__ atx,          // ws: [B][K][C]
                float* __restrict__ asum)         // ws: [B][K]
{
    __shared__ float Xs[Csz * XS_STRIDE];   // X tile, [c][n]
    __shared__ float Cs[Ksz * CS_STRIDE];   // codewords, [k][c]
    __shared__ float As[Ksz * AS_STRIDE];   // softmax A tile, [k][n]
    __shared__ float x2s[TILE_N];
    __shared__ float c2s[Ksz];
    __shared__ float scs[Ksz];

    const int t    = threadIdx.x;
    const int lane = t & 31;
    const int wave = t >> 5;
    const int lo   = lane & 15;
    const int hi   = lane >> 4;

    const int b     = blockIdx.x >> 10;          // N/TILE_N = 1024 tiles per batch
    const int nbase = (blockIdx.x & 1023) << 7;

    // ---------------- stage tiles into LDS via async b128 (no VGPR bounce) --
    {
        const float* Xb = X + ((size_t)b * Csz) * (size_t)Nsz + nbase;
        // X: one c-row (128 floats = 512B) per wave-instruction; 8 rows/wave
        #pragma unroll
        for (int i = 0; i < 8; ++i) {
            const int c = wave * 8 + i;
            async_copy_b128(Xb + (size_t)c * Nsz + 4 * lane,
                            &Xs[c * XS_STRIDE + 4 * lane]);
        }
        // CW: two k-rows (2 x 64 floats) per wave-instruction; 2 instr/wave
        #pragma unroll
        for (int ii = 0; ii < 2; ++ii) {
            const int k = 2 * (wave * 2 + ii) + hi;   // lanes 0-15 -> row 2p, 16-31 -> 2p+1
            async_copy_b128(CW + k * Csz + 4 * lo,
                            &Cs[k * CS_STRIDE + 4 * lo]);
        }
        if (t < Ksz) scs[t] = scale[t];
        async_wait_all();                 // each wave drains its ASYNCcnt
    }
    __syncthreads();

    // ---------------- row norms |x|^2 and |c|^2 ----------------------------
    if (t < TILE_N) {
        float s = 0.0f;
        #pragma unroll 8
        for (int c = 0; c < Csz; ++c) { float v = Xs[c * XS_STRIDE + t]; s += v * v; }
        x2s[t] = s;
    } else if (t < TILE_N + Ksz) {
        int k = t - TILE_N;
        float s = 0.0f;
        #pragma unroll 8
        for (int c = 0; c < Csz; ++c) { float v = Cs[k * CS_STRIDE + c]; s += v * v; }
        c2s[k] = s;
    }
    __syncthreads();

    // ---------------- GEMM1: XC[n,k] = sum_c Xf[n,c]*CW[k,c] ---------------
    // Wave w owns n-rows [w*16, w*16+16), all 32 k columns (two 16x16 tiles).
    v8f acc0 = {};   // k = 0..15
    v8f acc1 = {};   // k = 16..31
    const int nrow = wave * 16 + lo;             // A-fragment row for this lane
#if HAS_WMMA_F32
    #pragma unroll
    for (int s = 0; s < 16; ++s) {
        const int c0 = 4 * s + 2 * hi;           // A: K = 2*hi + vgpr (ISA 16x4 f32 layout)
        v2f a;  a.x  = Xs[(c0 + 0) * XS_STRIDE + nrow];
                a.y  = Xs[(c0 + 1) * XS_STRIDE + nrow];
        v2f b0; b0.x = Cs[lo * CS_STRIDE + c0];
                b0.y = Cs[lo * CS_STRIDE + c0 + 1];
        v2f b1; b1.x = Cs[(16 + lo) * CS_STRIDE + c0];
                b1.y = Cs[(16 + lo) * CS_STRIDE + c0 + 1];
        acc0 = __builtin_amdgcn_wmma_f32_16x16x4_f32(false, a, false, b0, (short)0, acc0, false, false);
        acc1 = __builtin_amdgcn_wmma_f32_16x16x4_f32(false, a, false, b1, (short)0, acc1, false, false);
    }
#else
    // VALU fallback (compile-safe path; matches D-fragment layout)
    for (int r = 0; r < 8; ++r) {
        const int nl = wave * 16 + 8 * hi + r;
        float s0 = 0.0f, s1 = 0.0f;
        for (int c = 0; c < Csz; ++c) {
            float xv = Xs[c * XS_STRIDE + nl];
            s0 += xv * Cs[lo * CS_STRIDE + c];
            s1 += xv * Cs[(16 + lo) * CS_STRIDE + c];
        }
        acc0[r] = s0; acc1[r] = s1;
    }
#endif

    // ---------------- logits + softmax over K (16 lanes x 2 regs) ----------
    // D-fragment: lane holds column k=lo (acc0) / k=16+lo (acc1),
    // rows n = wave*16 + 8*hi + r.
    {
        const float ck0 = c2s[lo],      ck1 = c2s[16 + lo];
        const float sk0 = scs[lo],      sk1 = scs[16 + lo];
        #pragma unroll
        for (int r = 0; r < 8; ++r) {
            const int nl = wave * 16 + 8 * hi + r;
            const float x2 = x2s[nl];
            float l0 = sk0 * (x2 + ck0 - 2.0f * acc0[r]);
            float l1 = sk1 * (x2 + ck1 - 2.0f * acc1[r]);
            float m = fmaxf(l0, l1);
            m = fmaxf(m, __shfl_xor(m, 1));
            m = fmaxf(m, __shfl_xor(m, 2));
            m = fmaxf(m, __shfl_xor(m, 4));
            m = fmaxf(m, __shfl_xor(m, 8));
            float e0 = __expf(l0 - m);
            float e1 = __expf(l1 - m);
            float sum = e0 + e1;
            sum += __shfl_xor(sum, 1);
            sum += __shfl_xor(sum, 2);
            sum += __shfl_xor(sum, 4);
            sum += __shfl_xor(sum, 8);
            float inv = 1.0f / sum;
            As[lo * AS_STRIDE + nl]        = e0 * inv;
            As[(16 + lo) * AS_STRIDE + nl] = e1 * inv;
        }
    }
    __syncthreads();

    // ---------------- GEMM2: AtX[k,c] += sum_n A[n,k]*Xf[n,c] --------------
    // 8 waves cover the 32x64 output: wave -> (kt, ct) 16x16 tile.
    v8f acc2 = {};
    const int kt = (wave >> 2) * 16;
    const int ct = (wave & 3) * 16;
#if HAS_WMMA_F32
    #pragma unroll
    for (int s = 0; s < 32; ++s) {
        const int n0 = 4 * s + 2 * hi;
        v2f a;  a.x  = As[(kt + lo) * AS_STRIDE + n0];
                a.y  = As[(kt + lo) * AS_STRIDE + n0 + 1];
        v2f bb; bb.x = Xs[(ct + lo) * XS_STRIDE + n0];
                bb.y = Xs[(ct + lo) * XS_STRIDE + n0 + 1];
        acc2 = __builtin_amdgcn_wmma_f32_16x16x4_f32(false, a, false, bb, (short)0, acc2, false, false);
    }
#else
    for (int r = 0; r < 8; ++r) {
        float s2 = 0.0f;
        for (int n = 0; n < TILE_N; ++n)
            s2 += As[(kt + 8 * hi + r) * AS_STRIDE + n] * Xs[(ct + lo) * XS_STRIDE + n];
        acc2[r] = s2;
    }
#endif
    #pragma unroll
    for (int r = 0; r < 8; ++r) {
        atomicAdd(&atx[((size_t)b * Ksz + kt + 8 * hi + r) * Csz + ct + lo], acc2[r]);
    }

    // ---------------- Asum[k] partials -------------------------------------
    if (t < Ksz) {
        float s = 0.0f;
        #pragma unroll 8
        for (int n = 0; n < TILE_N; ++n) s += As[t * AS_STRIDE + n];
        atomicAdd(&asum[b * Ksz + t], s);
    }

    // ---------------- coefA store: streamed, non-temporal ------------------
    {
        float* coefA = out + (size_t)Bsz * Ksz * Csz;   // E occupies first 8192
        const int nl = t & 127;
        const int kh = (t >> 7) * 16;
        #pragma unroll
        for (int kk = 0; kk < 16; ++kk) {
            const int k = kh + kk;
            __builtin_nontemporal_store(
                As[k * AS_STRIDE + nl],
                &coefA[((size_t)(b * Ksz + k)) * (size_t)Nsz + nbase + nl]);
        }
    }
}

__global__ void enc3d_finalize(const float* __restrict__ atx,
                               const float* __restrict__ asum,
                               const float* __restrict__ CW,
                               float* __restrict__ out)
{
    int i = blockIdx.x * blockDim.x + threadIdx.x;   // B*K*C = 8192
    if (i < Bsz * Ksz * Csz) {
        int c  = i & (Csz - 1);
        int k  = (i >> 6) & (Ksz - 1);
        int bk = i >> 6;
        out[i] = atx[i] - asum[bk] * CW[k * Csz + c];
    }
}

extern "C" void kernel_launch(void* const* d_in, const int* in_sizes, int n_in,
                              void* d_out, int out_size, void* d_ws, size_t ws_size,
                              hipStream_t stream) {
    const float* X     = (const float*)d_in[0];
    const float* CW    = (const float*)d_in[1];
    const float* scale = (const float*)d_in[2];
    float* out  = (float*)d_out;
    float* ws   = (float*)d_ws;
    float* atx  = ws;                 // [B][K][C]
    float* asum = ws + WS_ATX;        // [B][K]

    enc3d_init_ws<<<(WS_TOTAL + 255) / 256, 256, 0, stream>>>(ws, WS_TOTAL);
    enc3d_main<<<Bsz * (Nsz / TILE_N), 256, 0, stream>>>(X, CW, scale, out, atx, asum);
    enc3d_finalize<<<(Bsz * Ksz * Csz + 255) / 256, 256, 0, stream>>>(atx, asum, CW, out);
}